// FilterPool_25116968747649
// MI455X (gfx1250) — compile-verified
//
#include <hip/hip_runtime.h>
#include <hip/hip_bf16.h>
#include <stdint.h>

// ---------------------------------------------------------------------------
// PrRoI pooling (FILTER_SIZE=4, FEATURE_STRIDE=16) on gfx1250.
//
// Fused-GEMM formulation per sample n:
//   out[c][pq] = sum_k feat[c][k] * B[k][pq],   k = h*72 + w,  pq = p*4+q
//   B[k][pq]   = gy[p][h] * gx[q][w]
// => M=256 (c), K=5184, N=16 (pq): maps exactly onto V_WMMA_F32_16X16X4_F32
//    with full M/N/K utilization.  feat (340 MB) is streamed exactly once ->
//    HBM-bound floor ~14.6 us @ 23.3 TB/s.
//
// Data movement: per wave, feat rows (16 ch x 72 cols = 4608 B) are staged
// into an LDS ring buffer with GLOBAL_LOAD_ASYNC_TO_LDS_B128 (9 per row,
// 3 rows / 27 ops in flight, tracked by ASYNCcnt), so the WMMA loop only
// touches low-latency LDS instead of serializing on global load latency.
//
// Lane layout (CDNA5 ISA 7.12.2, 32-bit 16x4 A; B mirrored):
//   A: VGPR v, lane l: M = l&15,  K = v + 2*(l>>4)
//   B: VGPR v, lane l: N = l&15,  K = v + 2*(l>>4)
//   D: VGPR r, lane l: N = l&15,  M = r + 8*(l>>4)
// ---------------------------------------------------------------------------

#define C_DIM      256
#define H_DIM      72
#define W_DIM      72
#define HW_DIM     (H_DIM * W_DIM)
#define WS_PER_N   578            // gy[288] | gx[288] | inv_area | pad
#define ROW_BYTES  (W_DIM * 4)    // 288 B per channel-row
#define TILE_CH    16
#define TILE_ROW_FLOATS (TILE_CH * W_DIM)      // 1152 floats per staged row
#define TILE_ROW_BYTES  (TILE_ROW_FLOATS * 4)  // 4608 B
#define NBUF       4
#define ASYNC_PER_ROW 9           // 4608 B / (32 lanes * 16 B)

typedef float v2f __attribute__((ext_vector_type(2)));
typedef float v8f __attribute__((ext_vector_type(8)));

__device__ __forceinline__ float Gfun(float u) {
    u = fminf(1.0f, fmaxf(-1.0f, u));
    if (u <= 0.0f) { float t = u + 1.0f; return 0.5f * t * t; }
    float t = 1.0f - u; return 1.0f - 0.5f * t * t;
}

// ---------------------------------------------------------------------------
// Kernel 1: per-ROI exact integration weights + inverse bin area.
// ws layout per n: [0..287]=gy[p*72+h], [288..575]=gx[q*72+w], [576]=inv_area
// ---------------------------------------------------------------------------
__global__ void prroi_weights_kernel(const float* __restrict__ bb,
                                     float* __restrict__ ws) {
    const int n = blockIdx.x;
    const float x = bb[n * 4 + 0];
    const float y = bb[n * 4 + 1];
    const float w = bb[n * 4 + 2];
    const float h = bb[n * 4 + 3];
    const float scale = 1.0f / 16.0f;
    const float x1 = x * scale, y1 = y * scale;
    const float x2 = (x + w) * scale, y2 = (y + h) * scale;
    const float binx = (x2 - x1) * 0.25f;
    const float biny = (y2 - y1) * 0.25f;

    float* base = ws + (size_t)n * WS_PER_N;
    for (int idx = threadIdx.x; idx < 577; idx += blockDim.x) {
        if (idx < 576) {
            const int isX = (idx >= 288);
            const int r   = isX ? (idx - 288) : idx;
            const int k   = r / 72;
            const int i   = r % 72;
            const float c1  = isX ? x1 : y1;
            const float bsz = isX ? binx : biny;
            const float a = c1 + (float)k * bsz;
            const float b = a + bsz;
            base[idx] = Gfun(b - (float)i) - Gfun(a - (float)i);
        } else {
            const float area = fmaxf(binx * biny, 0.0f);
            base[576] = (area > 0.0f) ? (1.0f / fmaxf(area, 1e-12f)) : 0.0f;
        }
    }
}

// ---------------------------------------------------------------------------
// Kernel 2: async-LDS staged, WMMA f32 fused GEMM.
// grid = (N, 8), block = 64 (2 waves). Wave -> one 16-channel tile of one n.
// ---------------------------------------------------------------------------
__global__ void __launch_bounds__(64)
prroi_wmma_async_kernel(const float* __restrict__ feat,
                        const float* __restrict__ ws,
                        float* __restrict__ out) {
    __shared__ float ldsw[WS_PER_N];
    __shared__ __align__(16) float fbuf[2][NBUF][TILE_ROW_FLOATS]; // 36 KB

    const int n = blockIdx.x;
    {
        const float* src = ws + (size_t)n * WS_PER_N;
        for (int i = threadIdx.x; i < 577; i += blockDim.x) ldsw[i] = src[i];
    }
    __syncthreads();

    const int wave = threadIdx.x >> 5;
    const int lane = threadIdx.x & 31;
    const int c0   = (blockIdx.y * 2 + wave) * TILE_CH;
    const int m    = lane & 15;      // A row (channel) / D col (pq)
    const int hi   = lane >> 4;      // lane half -> K offset +2
    const int p    = m >> 2;
    const int q    = m & 3;

    const float* gyrow = &ldsw[p * 72];
    const float* gxrow = &ldsw[288 + q * 72 + 2 * hi];
    const float  inv_area = ldsw[576];

    // Per-lane source/dest addresses for the 9 async b128 copies of one row.
    // Row-tile byte offset for (instr i, lane): off = i*512 + lane*16,
    // mapping to channel ch = off/288, column byte col = off%288.
    const uint64_t tile_base =
        (uint64_t)(uintptr_t)(feat + (size_t)(n * C_DIM + c0) * HW_DIM);
    uint64_t garr[ASYNC_PER_ROW];
    uint32_t larr[ASYNC_PER_ROW];
    const uint32_t fb0 =
        (uint32_t)(uintptr_t)(&fbuf[wave][0][0]) + (uint32_t)lane * 16u;
#pragma unroll
    for (int i = 0; i < ASYNC_PER_ROW; ++i) {
        const uint32_t off = (uint32_t)(i * 512 + lane * 16);
        const uint32_t ch  = off / ROW_BYTES;
        const uint32_t col = off % ROW_BYTES;
        garr[i] = tile_base + (uint64_t)ch * (HW_DIM * 4) + col;
        larr[i] = fb0 + (uint32_t)(i * 512);
    }

    // Queue one feat row (16ch x 72col) into LDS ring slot via async copies.
    auto issue_row = [&](int row, int slot) {
        const uint64_t rofs = (uint64_t)row * ROW_BYTES;
        const uint32_t sofs = (uint32_t)slot * TILE_ROW_BYTES;
#pragma unroll
        for (int i = 0; i < ASYNC_PER_ROW; ++i) {
            uint64_t ga = garr[i] + rofs;
            uint32_t la = larr[i] + sofs;
            asm volatile("global_load_async_to_lds_b128 %0, %1, off th:TH_LOAD_NT"
                         :: "v"(la), "v"(ga) : "memory");
        }
    };

    v8f acc0 = {};
    v8f acc1 = {};

    // 18 WMMA K-steps over one staged row, A-fragments from LDS.
    auto consume_row = [&](int h, int slot) {
        const float gyv = gyrow[h];
        const float* f  = &fbuf[wave][slot][m * 72 + 2 * hi];
#pragma unroll
        for (int j = 0; j < 18; j += 2) {
            v2f a2 = *(const v2f*)(f + 4 * j);
            v2f g2 = *(const v2f*)(gxrow + 4 * j);
            v2f b2 = { gyv * g2.x, gyv * g2.y };
            acc0 = __builtin_amdgcn_wmma_f32_16x16x4_f32(
                false, a2, false, b2, (short)0, acc0, false, false);
            v2f a3 = *(const v2f*)(f + 4 * j + 4);
            v2f g3 = *(const v2f*)(gxrow + 4 * j + 4);
            v2f b3 = { gyv * g3.x, gyv * g3.y };
            acc1 = __builtin_amdgcn_wmma_f32_16x16x4_f32(
                false, a3, false, b3, (short)0, acc1, false, false);
        }
    };

    // Software pipeline: 3 rows (27 async ops) in flight; async completions
    // are in-order, so asynccnt<=18 guarantees the oldest row has landed.
    issue_row(0, 0);
    issue_row(1, 1);
    issue_row(2, 2);
    for (int h = 0; h < H_DIM - 3; ++h) {
        asm volatile("s_wait_asynccnt 0x12" ::: "memory");  // row h ready
        consume_row(h, h & 3);
        issue_row(h + 3, (h + 3) & 3);                      // safe: slot (h-1)&3
    }
    asm volatile("s_wait_asynccnt 0x12" ::: "memory");
    consume_row(H_DIM - 3, (H_DIM - 3) & 3);
    asm volatile("s_wait_asynccnt 0x9" ::: "memory");
    consume_row(H_DIM - 2, (H_DIM - 2) & 3);
    asm volatile("s_wait_asynccnt 0x0" ::: "memory");
    consume_row(H_DIM - 1, (H_DIM - 1) & 3);

    // D layout: VGPR r, lane l -> M = r + 8*hi (channel), N = m (pq).
    float* orow = out + (size_t)(n * C_DIM + c0) * 16 + m;
#pragma unroll
    for (int r = 0; r < 8; ++r) {
        const int M = r + 8 * hi;
        orow[(size_t)M * 16] = (acc0[r] + acc1[r]) * inv_area;
    }
}

// ---------------------------------------------------------------------------
extern "C" void kernel_launch(void* const* d_in, const int* in_sizes, int n_in,
                              void* d_out, int out_size, void* d_ws, size_t ws_size,
                              hipStream_t stream) {
    const float* feat = (const float*)d_in[0];
    const float* bb   = (const float*)d_in[1];
    float* out = (float*)d_out;
    float* ws  = (float*)d_ws;

    const int n_rois = in_sizes[1] / 4;   // 64

    prroi_weights_kernel<<<dim3(n_rois), dim3(128), 0, stream>>>(bb, ws);
    prroi_wmma_async_kernel<<<dim3(n_rois, 8), dim3(64), 0, stream>>>(feat, ws, out);
}